// Tsne_23759759081540
// MI455X (gfx1250) — compile-verified
//
#include <hip/hip_runtime.h>
#include <hip/hip_bf16.h>

typedef float v2f __attribute__((ext_vector_type(2)));
typedef float v8f __attribute__((ext_vector_type(8)));

#define TSNE_EPS   1e-12f
#define TSNE_TOL   1e-5f
#define TSNE_MAXIT 150
#define LOG2E      1.4426950408889634f
#define BLK        256
#define NWAVE      (BLK / 32)

__device__ __forceinline__ float waveReduceSum(float v) {
    #pragma unroll
    for (int off = 16; off > 0; off >>= 1) v += __shfl_xor(v, off, 32);
    return v;
}

// ---------------------------------------------------------------- init: zero accumulators
__global__ void tsne_init_kernel(float* out, float* sumAff) {
    if (threadIdx.x == 0 && blockIdx.x == 0) { out[0] = 0.0f; sumAff[0] = 0.0f; }
}

// ---------------------------------------------------------------- row norms s[i] = ||X_i||^2
__global__ __launch_bounds__(BLK) void tsne_rownorm_kernel(const float* __restrict__ X,
                                                           float* __restrict__ s,
                                                           int n, int d) {
    __shared__ float red[NWAVE];
    const int i = blockIdx.x, t = threadIdx.x;
    const int lane = t & 31, wave = t >> 5;
    float acc = 0.0f;
    const float* row = X + (size_t)i * d;
    for (int k = t; k < d; k += BLK) { float x = row[k]; acc += x * x; }
    acc = waveReduceSum(acc);
    if (lane == 0) red[wave] = acc;
    __syncthreads();
    if (t == 0) {
        float v = 0.0f;
        #pragma unroll
        for (int w = 0; w < NWAVE; ++w) v += red[w];
        s[i] = v;
    }
}

// ---------------------------------------------------------------- D = clamp(s_i + s_j - 2 X X^T, 0)
// Each wave computes a 32x32 output block as a 2x2 grid of 16x16 WMMA tiles:
// 4 independent accumulator chains, 4 b64 loads -> 4 v_wmma_f32_16x16x4_f32 per k-step.
// A fragment (16x4 f32): m = lane&15, VGPR0 K=klo, VGPR1 K=klo+1, klo=(lane>=16)?2:0.
// C/D layout: N = lane&15, M = vgpr + (lane>=16 ? 8 : 0).
__global__ __launch_bounds__(BLK) void tsne_dist_wmma_kernel(const float* __restrict__ X,
                                                             const float* __restrict__ s,
                                                             float* __restrict__ D,
                                                             int n, int d) {
    const int lane = threadIdx.x & 31;
    const int wave = threadIdx.x >> 5;
    const int tilesPerRow = n >> 5;                 // 32-wide macro tiles
    const int tile = blockIdx.x * NWAVE + wave;     // wave-uniform -> EXEC stays all-1s
    if (tile >= tilesPerRow * tilesPerRow) return;
    const int bi = tile / tilesPerRow;
    const int bj = tile - bi * tilesPerRow;
    const int i0 = bi << 5, j0 = bj << 5;

    const int mr  = lane & 15;
    const int klo = (lane >> 4) << 1;               // 0 or 2
    const float* a0p = X + (size_t)(i0      + mr) * d + klo;
    const float* a1p = X + (size_t)(i0 + 16 + mr) * d + klo;
    const float* b0p = X + (size_t)(j0      + mr) * d + klo;
    const float* b1p = X + (size_t)(j0 + 16 + mr) * d + klo;

    v8f c00 = {0,0,0,0,0,0,0,0}, c01 = c00, c10 = c00, c11 = c00;
    #pragma unroll 2
    for (int k = 0; k < d; k += 4) {
        v2f a0 = *(const v2f*)(a0p + k);
        v2f a1 = *(const v2f*)(a1p + k);
        v2f b0 = *(const v2f*)(b0p + k);
        v2f b1 = *(const v2f*)(b1p + k);
        c00 = __builtin_amdgcn_wmma_f32_16x16x4_f32(false, a0, false, b0, (short)0, c00, false, false);
        c01 = __builtin_amdgcn_wmma_f32_16x16x4_f32(false, a0, false, b1, (short)0, c01, false, false);
        c10 = __builtin_amdgcn_wmma_f32_16x16x4_f32(false, a1, false, b0, (short)0, c10, false, false);
        c11 = __builtin_amdgcn_wmma_f32_16x16x4_f32(false, a1, false, b1, (short)0, c11, false, false);
    }

    const int Mo = (lane >> 4) << 3;                // 0 or 8
    const float sj0 = s[j0 + mr];
    const float sj1 = s[j0 + 16 + mr];
    #pragma unroll
    for (int v = 0; v < 8; ++v) {
        const int M0 = i0 + Mo + v;
        const int M1 = i0 + 16 + Mo + v;
        const float si0 = s[M0], si1 = s[M1];
        D[(size_t)M0 * n + (j0 + mr)]      = fmaxf(si0 + sj0 - 2.0f * c00[v], 0.0f);
        D[(size_t)M0 * n + (j0 + 16 + mr)] = fmaxf(si0 + sj1 - 2.0f * c01[v], 0.0f);
        D[(size_t)M1 * n + (j0 + mr)]      = fmaxf(si1 + sj0 - 2.0f * c10[v], 0.0f);
        D[(size_t)M1 * n + (j0 + 16 + mr)] = fmaxf(si1 + sj1 - 2.0f * c11[v], 0.0f);
    }
}

// ---------------------------------------------------------------- per-row perplexity bisection
// One block per row; row (pre-scaled by log2e) staged in LDS once.
// Single-pass entropy via H = log2(sumP) + (beta*sum(P*sd) - EPS*log2(EPS)) / sumP
// where sd = d*log2e, P = exp2(-beta*sd).  Joint-P row normalizer = sumP - EPS.
__global__ __launch_bounds__(BLK) void tsne_beta_kernel(const float* __restrict__ D,
                                                        float* __restrict__ beta_out,
                                                        float* __restrict__ rowsum_out,
                                                        int n) {
    __shared__ float row[4096];
    __shared__ float redP[NWAVE], redPD[NWAVE];
    const int i = blockIdx.x, t = threadIdx.x;
    const int lane = t & 31, wave = t >> 5;
    const float* Drow = D + (size_t)i * n;
    for (int j = t; j < n; j += BLK) row[j] = Drow[j] * LOG2E;
    __syncthreads();

    const float logU     = log2f(30.0f);
    const float epsTerm  = -TSNE_EPS * log2f(TSNE_EPS);   // diagonal's -Pn*log2P piece (pre-division)
    float beta = 1.0f, bmin = 1e-6f, bmax = 1e6f;
    float sumP = 0.0f;

    for (int iter = 0; iter <= TSNE_MAXIT; ++iter) {
        float ps = 0.0f, pd = 0.0f;
        for (int j = t; j < n; j += BLK) {
            const float sd = row[j];
            const float p  = (j == i) ? TSNE_EPS : exp2f(-beta * sd);
            ps += p;
            pd = fmaf(p, (j == i) ? 0.0f : sd, pd);
        }
        ps = waveReduceSum(ps);
        pd = waveReduceSum(pd);
        if (lane == 0) { redP[wave] = ps; redPD[wave] = pd; }
        __syncthreads();
        float sP = 0.0f, sPD = 0.0f;
        #pragma unroll
        for (int w = 0; w < NWAVE; ++w) { sP += redP[w]; sPD += redPD[w]; }
        __syncthreads();                                   // protect LDS for next iter

        sumP = fmaxf(sP, TSNE_EPS);
        const float H = log2f(sumP) + (beta * sPD + epsTerm) / sumP;
        const float Hdiff = H - logU;                      // uniform across block

        if (fabsf(Hdiff) <= TSNE_TOL || iter == TSNE_MAXIT) break;
        if (Hdiff > 0.0f) { bmin = beta; beta = (beta + bmax) * 0.5f; }
        else              { bmax = beta; beta = (beta + bmin) * 0.5f; }
    }

    if (t == 0) { beta_out[i] = beta; rowsum_out[i] = sumP - TSNE_EPS; }
}

// ---------------------------------------------------------------- sum of Student-t affinities
__global__ __launch_bounds__(BLK) void tsne_aff_kernel(const float* __restrict__ Y,
                                                       float* __restrict__ sumAff, int n) {
    __shared__ float red[NWAVE];
    const int i = blockIdx.x, t = threadIdx.x;
    const int lane = t & 31, wave = t >> 5;
    const float yi0 = Y[2 * i], yi1 = Y[2 * i + 1];
    float acc = 0.0f;
    for (int j = t; j < n; j += BLK) {
        if (j == i) continue;
        float d0 = yi0 - Y[2 * j], d1 = yi1 - Y[2 * j + 1];
        acc += 1.0f / (1.0f + d0 * d0 + d1 * d1);
    }
    acc = waveReduceSum(acc);
    if (lane == 0) red[wave] = acc;
    __syncthreads();
    if (t == 0) {
        float v = 0.0f;
        #pragma unroll
        for (int w = 0; w < NWAVE; ++w) v += red[w];
        atomicAdd(sumAff, v);
    }
}

// ---------------------------------------------------------------- fused KL divergence
// loss = sum_ij P_ij * log2(P_ij / Q_ij)
// P_ij = max((Pe_ij/rs_i + Pe_ji/rs_j)/(2n), EPS) * 12 ; Pe from symmetric D
// Q_ij = max(aff_ij / sumAff, EPS)
__global__ __launch_bounds__(BLK) void tsne_loss_kernel(const float* __restrict__ D,
                                                        const float* __restrict__ beta,
                                                        const float* __restrict__ rowsum,
                                                        const float* __restrict__ Y,
                                                        const float* __restrict__ sumAffPtr,
                                                        float* __restrict__ out, int n) {
    __shared__ float red[NWAVE];
    const int i = blockIdx.x, t = threadIdx.x;
    const int lane = t & 31, wave = t >> 5;
    const float bi  = beta[i] * LOG2E;
    const float ri  = 1.0f / fmaxf(rowsum[i], TSNE_EPS);
    const float yi0 = Y[2 * i], yi1 = Y[2 * i + 1];
    const float invSumAff = 1.0f / sumAffPtr[0];
    const float inv2n = 1.0f / (2.0f * (float)n);
    const float* Drow = D + (size_t)i * n;

    float acc = 0.0f;
    for (int j = t; j < n; j += BLK) {
        const float dij = Drow[j];
        float peij, peji, aff;
        if (j == i) { peij = 0.0f; peji = 0.0f; aff = 0.0f; }
        else {
            peij = exp2f(-bi * dij);
            peji = exp2f(-beta[j] * LOG2E * dij);
            float d0 = yi0 - Y[2 * j], d1 = yi1 - Y[2 * j + 1];
            aff = 1.0f / (1.0f + d0 * d0 + d1 * d1);
        }
        const float rj = 1.0f / fmaxf(rowsum[j], TSNE_EPS);
        const float pj = (peij * ri + peji * rj) * inv2n;
        const float P  = fmaxf(pj, TSNE_EPS) * 12.0f;
        const float Q  = fmaxf(aff * invSumAff, TSNE_EPS);
        acc += P * log2f(P / Q);
    }
    acc = waveReduceSum(acc);
    if (lane == 0) red[wave] = acc;
    __syncthreads();
    if (t == 0) {
        float v = 0.0f;
        #pragma unroll
        for (int w = 0; w < NWAVE; ++w) v += red[w];
        atomicAdd(out, v);
    }
}

// ---------------------------------------------------------------- launch
extern "C" void kernel_launch(void* const* d_in, const int* in_sizes, int n_in,
                              void* d_out, int out_size, void* d_ws, size_t ws_size,
                              hipStream_t stream) {
    const float* X = (const float*)d_in[0];
    const float* Y = (const float*)d_in[1];
    float* out = (float*)d_out;

    const int n = in_sizes[1] / 2;            // 4096
    const int d = in_sizes[0] / n;            // 512

    float* D       = (float*)d_ws;            // n*n
    float* s       = D + (size_t)n * n;       // n
    float* beta    = s + n;                   // n
    float* rowsum  = beta + n;                // n
    float* sumAff  = rowsum + n;              // 1

    tsne_init_kernel<<<1, 32, 0, stream>>>(out, sumAff);
    tsne_rownorm_kernel<<<n, BLK, 0, stream>>>(X, s, n, d);

    const int tilesPerRow = n >> 5;
    const int nTiles = tilesPerRow * tilesPerRow;
    tsne_dist_wmma_kernel<<<(nTiles + NWAVE - 1) / NWAVE, BLK, 0, stream>>>(X, s, D, n, d);

    tsne_beta_kernel<<<n, BLK, 0, stream>>>(D, beta, rowsum, n);
    tsne_aff_kernel<<<n, BLK, 0, stream>>>(Y, sumAff, n);
    tsne_loss_kernel<<<n, BLK, 0, stream>>>(D, beta, rowsum, Y, sumAff, out, n);
}